// HyperSurfaceLoss_21406117004040
// MI455X (gfx1250) — compile-verified
//
#include <hip/hip_runtime.h>
#include <hip/hip_bf16.h>

typedef __attribute__((ext_vector_type(2))) float v2f;
typedef __attribute__((ext_vector_type(4))) float v4f;
typedef __attribute__((ext_vector_type(8))) float v8f;
typedef __attribute__((ext_vector_type(4))) int v4i;

#if defined(__has_builtin)
#if __has_builtin(__builtin_amdgcn_global_load_async_to_lds_b128)
#define HAVE_ASYNC_LDS 1
#endif
#if __has_builtin(__builtin_amdgcn_s_wait_asynccnt)
#define HAVE_WAIT_ASYNC 1
#endif
#endif

typedef __attribute__((address_space(1))) v4i* gv4ip;  // global int4*
typedef __attribute__((address_space(3))) v4i* lv4ip;  // LDS int4*

// One cyclic-Jacobi rotation for pivot (p,q) of a symmetric 4x4.
#define JROT(app, aqq, apq, apr, aqr, aps, aqs)                          \
  do {                                                                   \
    float _o = (apq);                                                    \
    if (fabsf(_o) > 1e-12f) {                                            \
      float _tau = ((aqq) - (app)) / (2.0f * _o);                        \
      float _t = ((_tau >= 0.0f) ? 1.0f : -1.0f) /                       \
                 (fabsf(_tau) + sqrtf(1.0f + _tau * _tau));              \
      float _c = 1.0f / sqrtf(1.0f + _t * _t);                           \
      float _s = _t * _c;                                                \
      (app) -= _t * _o;                                                  \
      (aqq) += _t * _o;                                                  \
      (apq) = 0.0f;                                                      \
      float _x = (apr), _y = (aqr);                                      \
      (apr) = _c * _x - _s * _y;                                         \
      (aqr) = _s * _x + _c * _y;                                         \
      _x = (aps); _y = (aqs);                                            \
      (aps) = _c * _x - _s * _y;                                         \
      (aqs) = _s * _x + _c * _y;                                         \
    }                                                                    \
  } while (0)

__device__ __forceinline__ float lambda_min4(float a00, float a01, float a02, float a03,
                                             float a11, float a12, float a13,
                                             float a22, float a23, float a33) {
  for (int sweep = 0; sweep < 4; ++sweep) {
    JROT(a00, a11, a01, a02, a12, a03, a13);
    JROT(a00, a22, a02, a01, a12, a03, a23);
    JROT(a00, a33, a03, a01, a13, a02, a23);
    JROT(a11, a22, a12, a01, a02, a13, a23);
    JROT(a11, a33, a13, a01, a03, a12, a23);
    JROT(a22, a33, a23, a02, a03, a12, a13);
  }
  return fminf(fminf(a00, a11), fminf(a22, a33));
}

// One wave32 per block; each block processes a tile of 32 rows of df (4x32 each).
// Staging: GLOBAL_LOAD_ASYNC_TO_LDS_B128 (ASYNCcnt path, no VGPR bounce).
// Grams: V_WMMA_F32_16X16X4_F32, 4 problems per 16x16 tile, K over 8 chunks of 4;
// symmetric input => the same VGPR pair is both the A and B fragment.
// Eigen-solve (lambda_min of 4x4) is lane-parallel Jacobi.
__global__ __launch_bounds__(32) void hs_tile_kernel(const float* __restrict__ f,
                                                     const float* __restrict__ df,
                                                     float* __restrict__ partial,
                                                     int nrows, float inv_n) {
  __shared__ float lds_stage[32 * 128];  // 16 KB: 32 rows x (4x32) f32
  __shared__ float lds_gram[32 * 16];    // 2 KB: 32 Gram matrices (col-major)

  const int lane = threadIdx.x;
  const int tile = blockIdx.x;
  const long tileBase = (long)tile * 32;
  const int nvalid = min(32, nrows - (int)tileBase);

  float lmin;
  if (nvalid == 32) {
    // ---- Stage 16 KB tile into LDS (coalesced 16B per lane) ----
    const float* src = df + tileBase * 128;
#ifdef HAVE_ASYNC_LDS
#pragma unroll 8
    for (int j = 0; j < 32; ++j) {
      gv4ip gsrc = (gv4ip)(src + (size_t)(j * 32 + lane) * 4);
      lv4ip ldst = (lv4ip)&lds_stage[(j * 32 + lane) * 4];
      __builtin_amdgcn_global_load_async_to_lds_b128(gsrc, ldst, 0, 0);
    }
#ifdef HAVE_WAIT_ASYNC
    __builtin_amdgcn_s_wait_asynccnt(0);
#else
    asm volatile("s_wait_asynccnt 0x0" ::: "memory");
#endif
#else
#pragma unroll 4
    for (int j = 0; j < 32; ++j) {
      v4f d = *(const v4f*)(src + (size_t)(j * 32 + lane) * 4);
      *(v4f*)&lds_stage[(j * 32 + lane) * 4] = d;
    }
#endif
    __syncthreads();

    // Lane roles for the WMMA fragments (ISA 7.12.2, 32-bit A 16x4):
    // lanes 0-15 hold A[m, K=0..1], lanes 16-31 hold A[m, K=2..3].
    const int m = lane & 15;   // fragment row (= 4*problem + subrow)
    const int hi = lane >> 4;  // K-half selector
    const int pin = m >> 2;    // problem within chain (0..3)
    const int r = m & 3;       // row within 4x32 block
    // D ownership: lane owns column n=lane&15 of rows 0-7 (lanes<16) / 8-15.
    const int n = m;
    const int pb = n >> 2;  // diagonal block this column belongs to
    const bool valid = hi ? (pb >= 2) : (pb < 2);
    const bool hiBlk = (pb & 1) != 0;

#pragma unroll
    for (int g = 0; g < 8; ++g) {  // 8 chains x 4 problems = 32 rows
      const int rowl = g * 4 + pin;
      const int abase = rowl * 128 + r * 32 + 2 * hi;
      v8f acc = {};
#pragma unroll
      for (int c = 0; c < 8; ++c) {  // K = 32 in chunks of 4
        v2f a = *(const v2f*)&lds_stage[abase + c * 4];
        // G += A_chunk * A_chunk^T : same fragment as A and B.
        acc = __builtin_amdgcn_wmma_f32_16x16x4_f32(
            false, a, false, a, (short)0, acc, false, false);
      }
      // Scatter this chain's diagonal 4x4 blocks (the Grams) to LDS.
      if (valid) {
        float c0 = hiBlk ? acc[4] : acc[0];
        float c1 = hiBlk ? acc[5] : acc[1];
        float c2 = hiBlk ? acc[6] : acc[2];
        float c3 = hiBlk ? acc[7] : acc[3];
        const int P = g * 4 + pb;  // problem index within tile (0..31)
        v4f col = {c0, c1, c2, c3};
        *(v4f*)&lds_gram[P * 16 + (n & 3) * 4] = col;
      }
    }
    __syncthreads();

    // Each lane now owns one problem's Gram (column-major, symmetric).
    const float* gp = &lds_gram[lane * 16];
    lmin = lambda_min4(gp[0], gp[1], gp[2], gp[3],
                       gp[5], gp[6], gp[7],
                       gp[10], gp[11],
                       gp[15]);
  } else {
    // Tail tile (not hit for N % 32 == 0): per-lane VALU Gram from global.
    float a00 = 0, a01 = 0, a02 = 0, a03 = 0, a11 = 0, a12 = 0, a13 = 0,
          a22 = 0, a23 = 0, a33 = 0;
    const long row = tileBase + lane;
    if (row < nrows) {
      const float* dr = df + row * 128;
      for (int k = 0; k < 32; ++k) {
        float x0 = dr[k], x1 = dr[32 + k], x2 = dr[64 + k], x3 = dr[96 + k];
        a00 += x0 * x0; a01 += x0 * x1; a02 += x0 * x2; a03 += x0 * x3;
        a11 += x1 * x1; a12 += x1 * x2; a13 += x1 * x3;
        a22 += x2 * x2; a23 += x2 * x3; a33 += x3 * x3;
      }
    }
    lmin = lambda_min4(a00, a01, a02, a03, a11, a12, a13, a22, a23, a33);
  }

  // Per-row contribution: |f_row|^2 / N  +  exp(-sigma_min^2) = exp(-lambda_min).
  float term = 0.0f;
  const long row = tileBase + lane;
  if (row < nrows) {
    v4f fv = *(const v4f*)(f + row * 4);
    float s = fv.x * fv.x + fv.y * fv.y + fv.z * fv.z + fv.w * fv.w;
    term = s * inv_n + expf(-lmin);
  }

  // Deterministic wave32 tree reduction.
#pragma unroll
  for (int o = 16; o > 0; o >>= 1) term += __shfl_down(term, o, 32);
  if (lane == 0) partial[tile] = term;
}

__global__ __launch_bounds__(512) void hs_reduce_kernel(const float* __restrict__ partial,
                                                        int n, float* __restrict__ out) {
  __shared__ float sm[512];
  float s = 0.0f;
  for (int i = threadIdx.x; i < n; i += 512) s += partial[i];
  sm[threadIdx.x] = s;
  __syncthreads();
  for (int k = 256; k > 0; k >>= 1) {
    if ((int)threadIdx.x < k) sm[threadIdx.x] += sm[threadIdx.x + k];
    __syncthreads();
  }
  if (threadIdx.x == 0) out[0] = sm[0];
}

extern "C" void kernel_launch(void* const* d_in, const int* in_sizes, int n_in,
                              void* d_out, int out_size, void* d_ws, size_t ws_size,
                              hipStream_t stream) {
  const float* f = (const float*)d_in[0];   // (N, 4) f32
  const float* df = (const float*)d_in[1];  // (N, 4, 32) f32
  const int nrows = in_sizes[0] / 4;        // 500000
  const int tiles = (nrows + 31) / 32;      // 15625
  float* partial = (float*)d_ws;            // tiles floats of scratch

  hs_tile_kernel<<<tiles, 32, 0, stream>>>(f, df, partial, nrows,
                                           1.0f / (float)nrows);
  hs_reduce_kernel<<<1, 512, 0, stream>>>(partial, tiles, (float*)d_out);
}